// MultiHopAggregator_27092653703580
// MI455X (gfx1250) — compile-verified
//
#include <hip/hip_runtime.h>
#include <hip/hip_bf16.h>

// ---------------------------------------------------------------------------
// MultiHopAggregator for MI455X (gfx1250, wave32, WMMA)
//
// out[e] = sum_j (adj[src[e],j] * struct[j])^2
//        = d[src[e]],  d[i] = sum_j adj[i,j]^2 * t[j],  t[j] = struct[j]^2
//
// Dominant cost: streaming 256 MiB of adj once (~11.5 us @ 23.3 TB/s).
// The matvec reduction runs through V_WMMA_F32_16X16X4_F32 with the t-vector
// broadcast across the 16 B-columns (redundant compute is free: AI=0.5 F/B).
// ---------------------------------------------------------------------------

#define N_NODES 8192
#define NNZ_CNT 262144
#define E_CNT   32768
#define PSI     64
#define COLS_PER_WG 256   // K-chunk per wave; grid = 512 row-tiles x 32 chunks

typedef __attribute__((ext_vector_type(2))) float v2f;
typedef __attribute__((ext_vector_type(8))) float v8f;

// --- zero the workspace (agg, t, d) ----------------------------------------
__global__ void zero_ws_kernel(float* __restrict__ ws, int n) {
  int i = blockIdx.x * blockDim.x + threadIdx.x;
  if (i < n) ws[i] = 0.0f;
}

// --- stage 1: edge MLP (1->64->ReLU->1) + scatter-add by col ---------------
__global__ __launch_bounds__(256) void edge_mlp_scatter_kernel(
    const float* __restrict__ values, const int* __restrict__ col,
    const float* __restrict__ w1e, const float* __restrict__ b1e,
    const float* __restrict__ w2e, const float* __restrict__ b2e,
    float* __restrict__ agg) {
  __shared__ float sw1[PSI], sb1[PSI], sw2[PSI];
  int t = threadIdx.x;
  if (t < PSI) { sw1[t] = w1e[t]; sb1[t] = b1e[t]; sw2[t] = w2e[t]; }
  __syncthreads();
  int i = blockIdx.x * blockDim.x + t;
  if (i >= NNZ_CNT) return;
  float v   = values[i];
  float acc = b2e[0];
#pragma unroll
  for (int p = 0; p < PSI; ++p) {
    float h = fmaf(v, sw1[p], sb1[p]);
    h = h > 0.0f ? h : 0.0f;
    acc = fmaf(h, sw2[p], acc);
  }
  atomicAdd(&agg[col[i]], acc);
}

// --- stage 2: node MLP, then square -> t[i] = struct[i]^2 ------------------
__global__ __launch_bounds__(256) void node_mlp_kernel(
    const float* __restrict__ agg,
    const float* __restrict__ w1n, const float* __restrict__ b1n,
    const float* __restrict__ w2n, const float* __restrict__ b2n,
    float* __restrict__ tsq) {
  __shared__ float sw1[PSI], sb1[PSI], sw2[PSI];
  int t = threadIdx.x;
  if (t < PSI) { sw1[t] = w1n[t]; sb1[t] = b1n[t]; sw2[t] = w2n[t]; }
  __syncthreads();
  int i = blockIdx.x * blockDim.x + t;
  if (i >= N_NODES) return;
  float x   = agg[i];
  float acc = b2n[0];
#pragma unroll
  for (int p = 0; p < PSI; ++p) {
    float h = fmaf(x, sw1[p], sb1[p]);
    h = h > 0.0f ? h : 0.0f;
    acc = fmaf(h, sw2[p], acc);
  }
  tsq[i] = acc * acc;
}

// --- stage 3: d[i] = sum_j adj[i,j]^2 * t[j] via WMMA f32 16x16x4 ----------
// One wave per (16-row tile, 256-col chunk).
// A operand (ISA layout): lane l -> row (l&15), K pair = (l>>4)*2  => each
// lane streams an aligned float2 of adj (nontemporal: 256MiB > 192MB L2),
// squares it in VALU. B operand: t broadcast across all 16 N-columns with
// the mirrored K indexing, so D[m][n] == partial d for row m, any n.
__global__ __launch_bounds__(32) void matvec_wmma_kernel(
    const float* __restrict__ adj, const float* __restrict__ tsq,
    float* __restrict__ dvec) {
  const int lane  = threadIdx.x;      // 0..31 (wave32)
  const int m     = lane & 15;        // row within tile
  const int khalf = lane >> 4;        // 0 -> K={0,1}, 1 -> K={2,3}

  const int row  = blockIdx.x * 16 + m;
  const int col0 = blockIdx.y * COLS_PER_WG;

  const float* __restrict__ ap = adj + (size_t)row * N_NODES + col0 + khalf * 2;
  const float* __restrict__ bp = tsq + col0 + khalf * 2;

  v8f c0 = {};
  v8f c1 = {};
#pragma unroll 4
  for (int k = 0; k < COLS_PER_WG; k += 8) {
    v2f a0 = __builtin_nontemporal_load((const v2f*)(ap + k));
    v2f a1 = __builtin_nontemporal_load((const v2f*)(ap + k + 4));
    v2f b0 = *(const v2f*)(bp + k);
    v2f b1 = *(const v2f*)(bp + k + 4);
    a0 *= a0;                         // adj^2 on the fly
    a1 *= a1;
    // 8 args: (neg_a, A, neg_b, B, c_mod, C, reuse_a, reuse_b)
    c0 = __builtin_amdgcn_wmma_f32_16x16x4_f32(
        false, a0, false, b0, (short)0, c0, false, false);
    c1 = __builtin_amdgcn_wmma_f32_16x16x4_f32(
        false, a1, false, b1, (short)0, c1, false, false);
  }

  // D layout: VGPR r, lane l -> M = r + 8*(l>>4), N = l&15; result is
  // replicated over N, so lanes with N==0 publish 8 rows each.
  if (m == 0) {
    const int rbase = blockIdx.x * 16 + khalf * 8;
#pragma unroll
    for (int r = 0; r < 8; ++r)
      atomicAdd(&dvec[rbase + r], c0[r] + c1[r]);
  }
}

// --- stage 4: gather out[e] = d[src[e]] ------------------------------------
__global__ __launch_bounds__(256) void gather_kernel(
    const int* __restrict__ src, const float* __restrict__ dvec,
    float* __restrict__ out) {
  int e = blockIdx.x * blockDim.x + threadIdx.x;
  if (e < E_CNT) out[e] = dvec[src[e]];
}

extern "C" void kernel_launch(void* const* d_in, const int* in_sizes, int n_in,
                              void* d_out, int out_size, void* d_ws, size_t ws_size,
                              hipStream_t stream) {
  const int*   col    = (const int*)  d_in[0];
  const float* values = (const float*)d_in[1];
  const float* adj    = (const float*)d_in[2];
  const int*   src    = (const int*)  d_in[3];
  const float* w1e    = (const float*)d_in[4];
  const float* b1e    = (const float*)d_in[5];
  const float* w2e    = (const float*)d_in[6];
  const float* b2e    = (const float*)d_in[7];
  const float* w1n    = (const float*)d_in[8];
  const float* b1n    = (const float*)d_in[9];
  const float* w2n    = (const float*)d_in[10];
  const float* b2n    = (const float*)d_in[11];
  float* out = (float*)d_out;

  float* ws   = (float*)d_ws;
  float* agg  = ws;                  // [N]
  float* tsq  = ws + N_NODES;        // [N]
  float* dvec = ws + 2 * N_NODES;    // [N]

  zero_ws_kernel<<<(3 * N_NODES + 255) / 256, 256, 0, stream>>>(ws, 3 * N_NODES);

  edge_mlp_scatter_kernel<<<NNZ_CNT / 256, 256, 0, stream>>>(
      values, col, w1e, b1e, w2e, b2e, agg);

  node_mlp_kernel<<<N_NODES / 256, 256, 0, stream>>>(
      agg, w1n, b1n, w2n, b2n, tsq);

  dim3 grid(N_NODES / 16, N_NODES / COLS_PER_WG);  // 512 x 32 waves
  matvec_wmma_kernel<<<grid, 32, 0, stream>>>(adj, tsq, dvec);

  gather_kernel<<<E_CNT / 256, 256, 0, stream>>>(src, dvec, out);
}